// NetGAT_69569880261287
// MI455X (gfx1250) — compile-verified
//
#include <hip/hip_runtime.h>
#include <hip/hip_bf16.h>

typedef __attribute__((ext_vector_type(2))) float v2f;
typedef __attribute__((ext_vector_type(8))) float v8f;

#define HEADS 6
#define NEG_SLOPE 0.2f

__device__ __forceinline__ float selu_f(float x) {
    const float scale = 1.0507009873554805f;
    const float alpha = 1.6732632423543772f;
    return x > 0.f ? scale * x : scale * alpha * (expf(x) - 1.f);
}

__device__ __forceinline__ float lrelu_f(float x) {
    return x > 0.f ? x : NEG_SLOPE * x;
}

// C[nrows x F] = act(A @ W^T + bias).  A is [>=tilesM*64 x lda] fully readable
// (zero-padded), W is [Fpad x K] row-major zero-padded, K % 4 == 0, Fpad % 16 == 0.
// One wave computes a 64x16 tile (4 M-tiles), B-frag reused across the 4 WMMAs.
// f32 WMMA 16x16x4 layouts per ISA 7.12.2: A-frag lanes 0-15 hold K=k0,k0+1,
// lanes 16-31 hold K=k0+2,k0+3; B mirrors with N=lane; D VGPR v = row M=v / v+8.
template <int K, int MT>
__global__ void gat_gemm_wmma(const float* __restrict__ A, int lda,
                              const float* __restrict__ W,
                              const float* __restrict__ bias,
                              float* __restrict__ C, int ldc,
                              int nrows, int Fpad, int F, int act, int tilesM) {
    int tilesN = Fpad >> 4;
    int wave = (blockIdx.x * blockDim.x + threadIdx.x) >> 5;
    if (wave >= tilesM * tilesN) return;          // wave-uniform exit
    int tm = wave / tilesN;
    int tn = wave % tilesN;
    int lane = threadIdx.x & 31;
    int hi   = lane >> 4;
    int l    = lane & 15;
    int col  = tn * 16 + l;                       // < Fpad always (W padded)
    int row0 = tm * (16 * MT) + l;

    const float* Wp = W + (size_t)col * K;
    v8f acc[MT];
#pragma unroll
    for (int m = 0; m < MT; ++m) acc[m] = (v8f){};

#pragma unroll
    for (int k0 = 0; k0 < K; k0 += 4) {
        int ka = k0 + hi * 2;                     // even -> 8B aligned loads
        v2f bv = *reinterpret_cast<const v2f*>(Wp + ka);
#pragma unroll
        for (int m = 0; m < MT; ++m) {
            v2f av = *reinterpret_cast<const v2f*>(A + (size_t)(row0 + m * 16) * lda + ka);
            acc[m] = __builtin_amdgcn_wmma_f32_16x16x4_f32(
                false, av, false, bv, (short)0, acc[m], false, false);
        }
    }

    if (col < F) {
        float bb = bias ? bias[col] : 0.f;
#pragma unroll
        for (int m = 0; m < MT; ++m) {
#pragma unroll
            for (int v = 0; v < 8; ++v) {
                int r = tm * (16 * MT) + m * 16 + hi * 8 + v;
                if (r < nrows) {
                    float xv = acc[m][v] + bb;
                    if (act) xv = selu_f(xv);
                    C[(size_t)r * ldc + col] = xv;
                }
            }
        }
    }
}

// Zero-padded copy: wpad[prows x pcols] <- w[rows x cols]
__global__ void gat_pad(const float* __restrict__ w, float* __restrict__ wpad,
                        int rows, int cols, int prows, int pcols) {
    int idx = blockIdx.x * blockDim.x + threadIdx.x;
    if (idx >= prows * pcols) return;
    int r = idx / pcols, c = idx % pcols;
    wpad[idx] = (r < rows && c < cols) ? w[(size_t)r * cols + c] : 0.f;
}

// al_s[n,h] = <proj[n,h,:], a_src[h,:]> ; al_d likewise. One thread per (n,h).
__global__ void gat_attn_coef(const float* __restrict__ proj,
                              const float* __restrict__ a_src,
                              const float* __restrict__ a_dst,
                              float* __restrict__ al_s, float* __restrict__ al_d,
                              int n, int O, int F) {
    int idx = blockIdx.x * blockDim.x + threadIdx.x;
    if (idx >= n * HEADS) return;
    int node = idx / HEADS, h = idx % HEADS;
    const float* p  = proj + (size_t)node * F + h * O;
    const float* as = a_src + h * O;
    const float* ad = a_dst + h * O;
    float ss = 0.f, dd = 0.f;
    for (int o = 0; o < O; ++o) { float v = p[o]; ss += v * as[o]; dd += v * ad[o]; }
    al_s[idx] = ss;
    al_d[idx] = dd;
}

__global__ void gat_fill(float* __restrict__ p, float v, int n) {
    int i = blockIdx.x * blockDim.x + threadIdx.x;
    if (i < n) p[i] = v;
}

// Pass 1: per-(dst,head) running max of leaky_relu(al_s[src]+al_d[dst]).
__global__ void gat_edge_max(const int* __restrict__ ei, int E, int n,
                             const float* __restrict__ al_s,
                             const float* __restrict__ al_d,
                             float* __restrict__ mbuf) {
    int e = blockIdx.x * blockDim.x + threadIdx.x;
    if (e >= E + n) return;
    int s, d;
    if (e < E) { s = ei[e]; d = ei[E + e]; } else { s = d = e - E; }
#pragma unroll
    for (int h = 0; h < HEADS; ++h) {
        float v = lrelu_f(al_s[s * HEADS + h] + al_d[d * HEADS + h]);
        atomicMax(&mbuf[d * HEADS + h], v);
    }
}

// Pass 2: per-(dst,head) sum of exp(e - max).
__global__ void gat_edge_sum(const int* __restrict__ ei, int E, int n,
                             const float* __restrict__ al_s,
                             const float* __restrict__ al_d,
                             const float* __restrict__ mbuf,
                             float* __restrict__ sbuf) {
    int e = blockIdx.x * blockDim.x + threadIdx.x;
    if (e >= E + n) return;
    int s, d;
    if (e < E) { s = ei[e]; d = ei[E + e]; } else { s = d = e - E; }
#pragma unroll
    for (int h = 0; h < HEADS; ++h) {
        float v = lrelu_f(al_s[s * HEADS + h] + al_d[d * HEADS + h]);
        atomicAdd(&sbuf[d * HEADS + h], expf(v - mbuf[d * HEADS + h]));
    }
}

// Pass 3: agg[dst,:] += alpha * proj[src,:]. One wave per edge, lanes stride F.
__global__ void gat_edge_agg(const int* __restrict__ ei, int E, int n,
                             const float* __restrict__ al_s,
                             const float* __restrict__ al_d,
                             const float* __restrict__ mbuf,
                             const float* __restrict__ sbuf,
                             const float* __restrict__ proj,
                             float* __restrict__ agg, int O, int F) {
    int wid  = (blockIdx.x * blockDim.x + threadIdx.x) >> 5;
    int lane = threadIdx.x & 31;
    if (wid >= E + n) return;
    int s, d;
    if (wid < E) { s = ei[wid]; d = ei[E + wid]; } else { s = d = wid - E; }
    for (int f = lane; f < F; f += 32) {
        int h = f / O;
        float ev    = lrelu_f(al_s[s * HEADS + h] + al_d[d * HEADS + h]);
        float alpha = expf(ev - mbuf[d * HEADS + h]) / sbuf[d * HEADS + h];
        atomicAdd(&agg[(size_t)d * F + f], proj[(size_t)s * F + f] * alpha);
    }
}

// out[n,o] = selu(mean_h agg[n,h,o] + bias[o])
__global__ void gat_finalize(const float* __restrict__ agg,
                             const float* __restrict__ bias,
                             float* __restrict__ outp,
                             int n, int O, int F, int ldc) {
    int idx = blockIdx.x * blockDim.x + threadIdx.x;
    if (idx >= n * O) return;
    int node = idx / O, o = idx % O;
    float acc = 0.f;
#pragma unroll
    for (int h = 0; h < HEADS; ++h) acc += agg[(size_t)node * F + h * O + o];
    outp[(size_t)node * ldc + o] = selu_f(acc * (1.f / HEADS) + bias[o]);
}

static inline int cdiv_i(long long a, long long b) { return (int)((a + b - 1) / b); }

extern "C" void kernel_launch(void* const* d_in, const int* in_sizes, int n_in,
                              void* d_out, int out_size, void* d_ws, size_t ws_size,
                              hipStream_t stream) {
    const float* x     = (const float*)d_in[0];
    const int*   ei    = (const int*)d_in[1];
    const float* lin_w = (const float*)d_in[2];
    const float* lin_b = (const float*)d_in[3];
    const int N    = in_sizes[0] / 10;
    const int E    = in_sizes[1] / 2;
    const int Npad = cdiv_i(N, 64) * 64;          // 64-row GEMM macro-tiles

    float* ws    = (float*)d_ws;
    float* xpad  = ws;                              // Npad x 12
    float* h_in  = xpad + (size_t)Npad * 12;        // Npad x 32 (ld=32)
    float* proj  = h_in + (size_t)Npad * 32;        // Npad x 192
    float* agg   = proj + (size_t)Npad * 192;       // N x 192
    float* al_s  = agg  + (size_t)N * 192;          // N x 6
    float* al_d  = al_s + (size_t)N * HEADS;
    float* mbuf  = al_d + (size_t)N * HEADS;
    float* sbuf  = mbuf + (size_t)N * HEADS;
    float* wlin  = sbuf + (size_t)N * HEADS;        // 16 x 12 padded lin_w
    float* w4p   = wlin + 16 * 12;                  // 16 x 16 padded W4

    const int T = 256;
    const float NEG_INF = -__builtin_inff();
    const int tilesM = Npad / 64;                   // MT = 4

    // --- stage padded operands ---
    gat_pad<<<cdiv_i((long long)Npad * 12, T), T, 0, stream>>>(x, xpad, N, 10, Npad, 12);
    gat_pad<<<1, 256, 0, stream>>>(lin_w, wlin, 16, 10, 16, 12);
    gat_pad<<<1, 256, 0, stream>>>((const float*)d_in[16], w4p, 12, 16, 16, 16);
    // zero the pad rows of h_in so layer GEMMs read finite zeros there
    if (Npad > N)
        gat_fill<<<cdiv_i((long long)(Npad - N) * 32, T), T, 0, stream>>>(
            h_in + (size_t)N * 32, 0.f, (Npad - N) * 32);

    // h = selu(x @ lin_w^T + lin_b): K=12 (padded), Fpad=16, F=16, ld=32
    gat_gemm_wmma<12, 4><<<cdiv_i((long long)tilesM * 1 * 32, T), T, 0, stream>>>(
        xpad, 12, wlin, lin_b, h_in, 32, N, 16, 16, /*act=*/1, tilesM);

    const int dims[4][2] = {{16, 32}, {32, 32}, {32, 16}, {16, 2}};
    for (int L = 0; L < 4; ++L) {
        int O = dims[L][1], F = HEADS * O;
        int Fpad = (L == 3) ? 16 : F;
        const float* W   = (L == 3) ? w4p : (const float*)d_in[4 + L * 4 + 0];
        const float* asr = (const float*)d_in[4 + L * 4 + 1];
        const float* adr = (const float*)d_in[4 + L * 4 + 2];
        const float* b   = (const float*)d_in[4 + L * 4 + 3];

        // proj = h_in @ W^T   [N x F]
        long long waves = (long long)tilesM * (Fpad >> 4);
        if (dims[L][0] == 16)
            gat_gemm_wmma<16, 4><<<cdiv_i(waves * 32, T), T, 0, stream>>>(
                h_in, 32, W, nullptr, proj, F, N, Fpad, F, 0, tilesM);
        else
            gat_gemm_wmma<32, 4><<<cdiv_i(waves * 32, T), T, 0, stream>>>(
                h_in, 32, W, nullptr, proj, F, N, Fpad, F, 0, tilesM);

        gat_attn_coef<<<cdiv_i((long long)N * HEADS, T), T, 0, stream>>>(
            proj, asr, adr, al_s, al_d, N, O, F);

        gat_fill<<<cdiv_i((long long)N * HEADS, T), T, 0, stream>>>(mbuf, NEG_INF, N * HEADS);
        gat_fill<<<cdiv_i((long long)N * HEADS, T), T, 0, stream>>>(sbuf, 0.f, N * HEADS);
        gat_fill<<<cdiv_i((long long)N * F, T), T, 0, stream>>>(agg, 0.f, N * F);

        gat_edge_max<<<cdiv_i((long long)(E + N), T), T, 0, stream>>>(ei, E, N, al_s, al_d, mbuf);
        gat_edge_sum<<<cdiv_i((long long)(E + N), T), T, 0, stream>>>(ei, E, N, al_s, al_d, mbuf, sbuf);
        gat_edge_agg<<<cdiv_i((long long)(E + N) * 32, T), T, 0, stream>>>(
            ei, E, N, al_s, al_d, mbuf, sbuf, proj, agg, O, F);

        float* outp = (L == 3) ? (float*)d_out : h_in;
        int ldc     = (L == 3) ? O : 32;
        gat_finalize<<<cdiv_i((long long)N * O, T), T, 0, stream>>>(agg, b, outp, N, O, F, ldc);
    }
}